// RoIPool_64630667870342
// MI455X (gfx1250) — compile-verified
//
#include <hip/hip_runtime.h>
#include <hip/hip_bf16.h>

typedef float v2f __attribute__((ext_vector_type(2)));
typedef float v8f __attribute__((ext_vector_type(8)));

#define EPSV 1e-5f

// ---------------------------------------------------------------------------
// Kernel 1: RoI pool. Reads only x[:, :, 2:6, 1:6]. Writes pooled as a GEMM
// A-matrix: A1[(n*4 + s)*256 + c], s = h*2 + w, 256 rows x 256 cols.
// ---------------------------------------------------------------------------
__global__ void roi_pool_kernel(const float* __restrict__ x, float* __restrict__ A1) {
    const int n = blockIdx.x;     // 64
    const int c = threadIdx.x;    // 256
    const float* base = x + ((size_t)(n * 256 + c) * 64) * 64;
    float v[4][5];
#pragma unroll
    for (int rr = 0; rr < 4; ++rr)
#pragma unroll
        for (int cc = 0; cc < 5; ++cc)
            v[rr][cc] = base[(2 + rr) * 64 + 1 + cc];
    float r0[5], r1[5];
#pragma unroll
    for (int cc = 0; cc < 5; ++cc) {
        r0[cc] = fmaxf(v[0][cc], v[1][cc]);
        r1[cc] = fmaxf(v[2][cc], v[3][cc]);
    }
    float p00 = fmaxf(fmaxf(r0[0], r0[1]), r0[2]);
    float p01 = fmaxf(fmaxf(r0[2], r0[3]), r0[4]);
    float p10 = fmaxf(fmaxf(r1[0], r1[1]), r1[2]);
    float p11 = fmaxf(fmaxf(r1[2], r1[3]), r1[4]);
    float* o = A1 + (size_t)n * 4 * 256 + c;
    o[0 * 256] = p00;   // s=0 (h=0,w=0)
    o[1 * 256] = p01;   // s=1 (h=0,w=1)
    o[2 * 256] = p10;   // s=2 (h=1,w=0)
    o[3 * 256] = p11;   // s=3 (h=1,w=1)
}

// ---------------------------------------------------------------------------
// Kernel 2: fused 1x1-conv (GEMM 256 x COUT, K=KIN) + BatchNorm + ReLU.
// One workgroup (8 waves) per head; WMMA accumulators stay in VGPRs; BN
// stats via LDS float atomics; barriers order GEMM -> stats -> writeback.
// RESHAPE=true writes the fc1 input layout: O[n*(COUT*4) + col*4 + s].
// ---------------------------------------------------------------------------
template <int KIN, int COUT, bool RESHAPE>
__global__ __launch_bounds__(256) void conv_bn_relu_kernel(
    const float* __restrict__ Ain, int a_head_stride,
    const float* __restrict__ Wt,          // (Hn, COUT, KIN)
    const float* __restrict__ g, const float* __restrict__ b,
    float* __restrict__ Out, int out_head_stride) {
    constexpr int NT  = COUT / 16;   // N tiles
    constexpr int TT  = 16 * NT;     // total 16x16 tiles (16 M-tiles)
    constexpr int TPW = TT / 8;      // tiles per wave (8 waves)

    __shared__ float lsum[COUT], lsq[COUT], lscale[COUT], lshift[COUT];

    const int h    = blockIdx.x;
    const int tid  = threadIdx.x;
    const int wave = tid >> 5;
    const int lane = tid & 31;
    const int half = lane >> 4;
    const int r    = lane & 15;

    if (tid < COUT) { lsum[tid] = 0.f; lsq[tid] = 0.f; }
    __syncthreads();

    const float* A = Ain + (size_t)h * a_head_stride;
    const float* W = Wt + (size_t)h * COUT * KIN;

    v8f acc[TPW];
#pragma unroll
    for (int i = 0; i < TPW; ++i) {
        const int tt = wave + 8 * i;
        const int mt = tt & 15, nt = tt >> 4;
        const int m0 = mt * 16, n0 = nt * 16;
        v8f c = {};
        const float2* arow = reinterpret_cast<const float2*>(A + (size_t)(m0 + r) * KIN);
        const float2* brow = reinterpret_cast<const float2*>(W + (size_t)(n0 + r) * KIN);
        for (int k0 = 0; k0 < KIN; k0 += 4) {
            float2 av = arow[(k0 >> 1) + half];
            float2 bv = brow[(k0 >> 1) + half];
            v2f a; a[0] = av.x; a[1] = av.y;
            v2f bb; bb[0] = bv.x; bb[1] = bv.y;
            c = __builtin_amdgcn_wmma_f32_16x16x4_f32(false, a, false, bb,
                                                      (short)0, c, false, false);
        }
        acc[i] = c;
        float s = 0.f, q = 0.f;
#pragma unroll
        for (int v = 0; v < 8; ++v) { float xv = c[v]; s += xv; q += xv * xv; }
        atomicAdd(&lsum[n0 + r], s);
        atomicAdd(&lsq[n0 + r], q);
    }
    __syncthreads();

    if (tid < COUT) {
        float mu  = lsum[tid] * (1.f / 256.f);
        float var = lsq[tid] * (1.f / 256.f) - mu * mu;
        float sc  = g[h * COUT + tid] * rsqrtf(var + EPSV);
        lscale[tid] = sc;
        lshift[tid] = b[h * COUT + tid] - mu * sc;
    }
    __syncthreads();

    float* O = Out + (size_t)h * out_head_stride;
#pragma unroll
    for (int i = 0; i < TPW; ++i) {
        const int tt = wave + 8 * i;
        const int mt = tt & 15, nt = tt >> 4;
        const int m0 = mt * 16, n0 = nt * 16;
        const int col = n0 + r;
        const float sc = lscale[col], sh = lshift[col];
#pragma unroll
        for (int v = 0; v < 8; ++v) {
            const int row = m0 + v + 8 * half;
            float val = acc[i][v] * sc + sh;
            val = fmaxf(val, 0.f);
            if (RESHAPE) {
                const int n = row >> 2, s = row & 3;
                O[n * (COUT * 4) + col * 4 + s] = val;
            } else {
                O[(size_t)row * COUT + col] = val;
            }
        }
    }
}

// ---------------------------------------------------------------------------
// Kernel 3: per-head GEMM (M x N, K) + bias (+ optional ReLU). Column-guarded
// for N not a multiple of 16 (fc2: N=100). Guards are lane selects so EXEC
// stays all-ones around WMMA.
// ---------------------------------------------------------------------------
template <int M, int N, int K, bool RELU>
__global__ __launch_bounds__(256) void gemm_bias_kernel(
    const float* __restrict__ Ain, int a_head_stride,
    const float* __restrict__ Wt,          // (Hn, N, K)
    const float* __restrict__ bias,        // (Hn, N)
    float* __restrict__ Out, int out_head_stride) {
    constexpr int MT = (M + 15) / 16, NT = (N + 15) / 16, TT = MT * NT;
    const int h    = blockIdx.y;
    const int tid  = threadIdx.x;
    const int wave = tid >> 5;
    const int lane = tid & 31;
    const int half = lane >> 4;
    const int r    = lane & 15;
    const int wglobal = blockIdx.x * (blockDim.x >> 5) + wave;
    const int wstride = gridDim.x * (blockDim.x >> 5);

    const float* A = Ain + (size_t)h * a_head_stride;
    const float* W = Wt + (size_t)h * N * K;
    float* O = Out + (size_t)h * out_head_stride;

    for (int tt = wglobal; tt < TT; tt += wstride) {
        const int mt = tt % MT, nt = tt / MT;
        const int m0 = mt * 16, n0 = nt * 16;
        const int col = n0 + r;
        const bool colok = (col < N);
        v8f c = {};
        const float2* arow = reinterpret_cast<const float2*>(A + (size_t)(m0 + r) * K);
        const float2* brow = reinterpret_cast<const float2*>(W + (size_t)(colok ? col : 0) * K);
        for (int k0 = 0; k0 < K; k0 += 4) {
            float2 av = arow[(k0 >> 1) + half];
            float2 bv = brow[(k0 >> 1) + half];
            v2f a; a[0] = av.x; a[1] = av.y;
            v2f bb; bb[0] = colok ? bv.x : 0.f; bb[1] = colok ? bv.y : 0.f;
            c = __builtin_amdgcn_wmma_f32_16x16x4_f32(false, a, false, bb,
                                                      (short)0, c, false, false);
        }
        if (colok) {
            const float bs = bias[h * N + col];
#pragma unroll
            for (int v = 0; v < 8; ++v) {
                const int row = m0 + v + 8 * half;
                float val = c[v] + bs;
                if (RELU) val = fmaxf(val, 0.f);
                O[(size_t)row * N + col] = val;
            }
        }
    }
}

// ---------------------------------------------------------------------------
// Kernel 4: mean over heads, final 100x100 matmul + bias, replicate rows x64.
// ---------------------------------------------------------------------------
__global__ void final_combine_kernel(const float* __restrict__ hout,  // [4][64][100]
                                     const float* __restrict__ W2,    // [100][100]
                                     const float* __restrict__ bias2, // [100]
                                     float* __restrict__ out) {       // [4096][100]
    __shared__ float comb[100];
    const int n = blockIdx.x;   // 64 distinct samples
    const int t = threadIdx.x;  // 128 threads
    if (t < 100) {
        float s = 0.f;
#pragma unroll
        for (int h = 0; h < 4; ++h) s += hout[((size_t)h * 64 + n) * 100 + t];
        comb[t] = s * 0.25f;
    }
    __syncthreads();
    if (t < 100) {
        float acc = bias2[t];
        for (int j = 0; j < 100; ++j) acc += comb[j] * W2[t * 100 + j];
        float* orow = out + (size_t)n * 64 * 100 + t;
#pragma unroll 4
        for (int rep = 0; rep < 64; ++rep) orow[(size_t)rep * 100] = acc;
    }
}

// ---------------------------------------------------------------------------
extern "C" void kernel_launch(void* const* d_in, const int* in_sizes, int n_in,
                              void* d_out, int out_size, void* d_ws, size_t ws_size,
                              hipStream_t stream) {
    const float* x     = (const float*)d_in[0];
    const float* w1    = (const float*)d_in[1];
    const float* g1    = (const float*)d_in[2];
    const float* b1    = (const float*)d_in[3];
    const float* w2    = (const float*)d_in[4];
    const float* g2    = (const float*)d_in[5];
    const float* b2    = (const float*)d_in[6];
    const float* w3    = (const float*)d_in[7];
    const float* g3    = (const float*)d_in[8];
    const float* b3    = (const float*)d_in[9];
    const float* lw1   = (const float*)d_in[10];
    const float* lb1   = (const float*)d_in[11];
    const float* lw2   = (const float*)d_in[12];
    const float* lb2   = (const float*)d_in[13];
    const float* W2    = (const float*)d_in[14];
    const float* bias2 = (const float*)d_in[15];
    float* out = (float*)d_out;

    float* ws   = (float*)d_ws;
    float* A1   = ws;                        // 256*256
    float* z1   = A1 + 256 * 256;            // 4*256*32
    float* z2   = z1 + 4 * 256 * 32;         // 4*256*64
    float* fcin = z2 + 4 * 256 * 64;         // 4*64*256
    float* h1   = fcin + 4 * 64 * 256;       // 4*64*512
    float* hout = h1 + 4 * 64 * 512;         // 4*64*100

    roi_pool_kernel<<<64, 256, 0, stream>>>(x, A1);

    conv_bn_relu_kernel<256, 32, false><<<4, 256, 0, stream>>>(
        A1, 0, w1, g1, b1, z1, 256 * 32);
    conv_bn_relu_kernel<32, 64, false><<<4, 256, 0, stream>>>(
        z1, 256 * 32, w2, g2, b2, z2, 256 * 64);
    conv_bn_relu_kernel<64, 64, true><<<4, 256, 0, stream>>>(
        z2, 256 * 64, w3, g3, b3, fcin, 64 * 256);

    gemm_bias_kernel<64, 512, 256, true><<<dim3(8, 4), 256, 0, stream>>>(
        fcin, 64 * 256, lw1, lb1, h1, 64 * 512);
    gemm_bias_kernel<64, 100, 512, false><<<dim3(4, 4), 256, 0, stream>>>(
        h1, 64 * 512, lw2, lb2, hout, 64 * 100);

    final_combine_kernel<<<64, 128, 0, stream>>>(hout, W2, bias2, out);
}